// RecPolicy_56839597195499
// MI455X (gfx1250) — compile-verified
//
#include <hip/hip_runtime.h>
#include <stdint.h>

#define RPB 256
#define LOG2E 1.4426950408889634f

// ---- CDNA5 transcendental helpers: prefer hardware v_tanh_f32 (gfx1250 TRANS op) ----
__device__ __forceinline__ float hw_tanh(float x) {
#if __has_builtin(__builtin_amdgcn_tanhf)
  return __builtin_amdgcn_tanhf(x);
#else
  // tanh(x) = 1 - 2/(1+exp2(2*log2(e)*x));  v_exp_f32 + v_rcp_f32
  float e = __builtin_amdgcn_exp2f(2.0f * LOG2E * x);
  return 1.0f - 2.0f * __builtin_amdgcn_rcpf(1.0f + e);
#endif
}

__device__ __forceinline__ float hw_sigmoid(float x) {
#if __has_builtin(__builtin_amdgcn_tanhf)
  // sigmoid(x) = 0.5 + 0.5*tanh(x/2): single TRANS op
  return __builtin_fmaf(0.5f, __builtin_amdgcn_tanhf(0.5f * x), 0.5f);
#else
  float e = __builtin_amdgcn_exp2f(-LOG2E * x);
  return __builtin_amdgcn_rcpf(1.0f + e);
#endif
}

// ---- CDNA5 async memory->LDS copies (ASYNCcnt path) ----
__device__ __forceinline__ void async_load_b128(const void* gsrc, const void* ldst) {
  uint32_t loff = (uint32_t)(uintptr_t)ldst;  // flat LDS addr keeps byte offset in [31:0]
  uint64_t ga   = (uint64_t)(uintptr_t)gsrc;
  asm volatile("global_load_async_to_lds_b128 %0, %1, off"
               :: "v"(loff), "v"(ga) : "memory");
}

__device__ __forceinline__ void async_load_b64(const void* gsrc, const void* ldst) {
  uint32_t loff = (uint32_t)(uintptr_t)ldst;
  uint64_t ga   = (uint64_t)(uintptr_t)gsrc;
  asm volatile("global_load_async_to_lds_b64 %0, %1, off"
               :: "v"(loff), "v"(ga) : "memory");
}

__device__ __forceinline__ void wait_async0() {
#if __has_builtin(__builtin_amdgcn_s_wait_asynccnt)
  __builtin_amdgcn_s_wait_asynccnt(0);
#else
  asm volatile("s_wait_asynccnt 0" ::: "memory");
#endif
}

// ---- GRU weights with pre-merged biases (r,z gates use b_ih+b_hh) ----
struct GruW {
  float wih[12], whh[12];
  float brz[4];  // merged b_ih+b_hh for rows 0..3 (r0,r1,z0,z1)
  float bin[2];  // b_ih rows 4,5 (n gate, input side)
  float bhn[2];  // b_hh rows 4,5 (n gate, hidden side; n = tanh(i_n + r*h_n))
};

__device__ __forceinline__ void gru_step(float x0, float x1, float& h0, float& h1,
                                         const GruW& W) {
  float gr0 = fmaf(W.wih[0], x0, fmaf(W.wih[1], x1, fmaf(W.whh[0], h0, fmaf(W.whh[1], h1, W.brz[0]))));
  float gr1 = fmaf(W.wih[2], x0, fmaf(W.wih[3], x1, fmaf(W.whh[2], h0, fmaf(W.whh[3], h1, W.brz[1]))));
  float gz0 = fmaf(W.wih[4], x0, fmaf(W.wih[5], x1, fmaf(W.whh[4], h0, fmaf(W.whh[5], h1, W.brz[2]))));
  float gz1 = fmaf(W.wih[6], x0, fmaf(W.wih[7], x1, fmaf(W.whh[6], h0, fmaf(W.whh[7], h1, W.brz[3]))));
  float in0 = fmaf(W.wih[8],  x0, fmaf(W.wih[9],  x1, W.bin[0]));
  float in1 = fmaf(W.wih[10], x0, fmaf(W.wih[11], x1, W.bin[1]));
  float hn0 = fmaf(W.whh[8],  h0, fmaf(W.whh[9],  h1, W.bhn[0]));
  float hn1 = fmaf(W.whh[10], h0, fmaf(W.whh[11], h1, W.bhn[1]));
  float r0 = hw_sigmoid(gr0), r1 = hw_sigmoid(gr1);
  float z0 = hw_sigmoid(gz0), z1 = hw_sigmoid(gz1);
  float n0 = hw_tanh(fmaf(r0, hn0, in0));
  float n1 = hw_tanh(fmaf(r1, hn1, in1));
  // (1-z)*n + z*h = n + z*(h-n)
  h0 = fmaf(z0, h0 - n0, n0);
  h1 = fmaf(z1, h1 - n1, n1);
}

__device__ __forceinline__ void compute_row(const float* rx, const GruW& up, const GruW& dn,
                                            const float* wobs, float bo0, float bo1,
                                            float wo0, float wo1, float bo, float* acts) {
  float obs0 = rx[0], obs1 = rx[1], obs2 = rx[2], obs3 = rx[3];
  float h0 = 0.0f, h1 = 0.0f;
  float hu0[7], hu1[7];
  #pragma unroll
  for (int k = 0; k < 7; ++k) {         // reversed(range(7)): i = 6..0
    const int i = 6 - k;
    gru_step(rx[4 + i], rx[11 + i], h0, h1, up);   // (j[i], jd[i])
    hu0[k] = h0; hu1[k] = h1;
  }
  // [obs(4), h(2)] @ w_obs.T + b_obs
  float m0 = fmaf(wobs[0], obs0, fmaf(wobs[1], obs1, fmaf(wobs[2], obs2,
             fmaf(wobs[3], obs3, fmaf(wobs[4], h0, fmaf(wobs[5], h1, bo0))))));
  float m1 = fmaf(wobs[6], obs0, fmaf(wobs[7], obs1, fmaf(wobs[8], obs2,
             fmaf(wobs[9], obs3, fmaf(wobs[10], h0, fmaf(wobs[11], h1, bo1))))));
  h0 = m0; h1 = m1;
  #pragma unroll
  for (int k = 0; k < 7; ++k) {
    gru_step(hu0[k], hu1[k], h0, h1, dn);
    acts[k] = fmaf(h1, wo1, fmaf(h0, wo0, bo));
  }
}

__global__ __launch_bounds__(RPB) void rec_policy_kernel(
    const float* __restrict__ x,
    const float* __restrict__ w_ih_up, const float* __restrict__ w_hh_up,
    const float* __restrict__ b_ih_up, const float* __restrict__ b_hh_up,
    const float* __restrict__ w_obs,   const float* __restrict__ b_obs,
    const float* __restrict__ w_ih_dn, const float* __restrict__ w_hh_dn,
    const float* __restrict__ b_ih_dn, const float* __restrict__ b_hh_dn,
    const float* __restrict__ w_out,   const float* __restrict__ b_out_p,
    float* __restrict__ out, int nrows) {
  __shared__ __align__(16) float tile [RPB * 18];  // 18432 B staged input (256 rows x 72 B)
  __shared__ __align__(16) float otile[RPB * 7];   //  7168 B output transpose buffer

  const int t = threadIdx.x;
  const long long row0 = (long long)blockIdx.x * RPB;
  const bool full = (row0 + RPB) <= (long long)nrows;   // uniform per block

  if (full) {
    // Coalesced async copy of 18432 B: 4 x b128 per thread (16384 B) + 1 x b64 (2048 B).
    // Block base = blockIdx*18432 (16B aligned); b64 tail at +16384 (8B aligned).
    const char* gsrc = (const char*)(x + row0 * 18);
    #pragma unroll
    for (int k = 0; k < 4; ++k) {
      const size_t off = (size_t)(t + k * RPB) * 16u;
      async_load_b128(gsrc + off, (const char*)tile + off);
    }
    {
      const size_t off = 16384u + (size_t)t * 8u;
      async_load_b64(gsrc + off, (const char*)tile + off);
    }
  }

  // Uniform weight loads (overlap with async copy); expected to scalarize.
  GruW up, dn;
  #pragma unroll
  for (int i = 0; i < 12; ++i) {
    up.wih[i] = w_ih_up[i]; up.whh[i] = w_hh_up[i];
    dn.wih[i] = w_ih_dn[i]; dn.whh[i] = w_hh_dn[i];
  }
  #pragma unroll
  for (int i = 0; i < 4; ++i) {
    up.brz[i] = b_ih_up[i] + b_hh_up[i];
    dn.brz[i] = b_ih_dn[i] + b_hh_dn[i];
  }
  #pragma unroll
  for (int i = 0; i < 2; ++i) {
    up.bin[i] = b_ih_up[4 + i]; up.bhn[i] = b_hh_up[4 + i];
    dn.bin[i] = b_ih_dn[4 + i]; dn.bhn[i] = b_hh_dn[4 + i];
  }
  float wobs[12];
  #pragma unroll
  for (int i = 0; i < 12; ++i) wobs[i] = w_obs[i];
  const float bo0 = b_obs[0], bo1 = b_obs[1];
  const float wo0 = w_out[0], wo1 = w_out[1], bo = b_out_p[0];

  if (full) {
    wait_async0();
    __syncthreads();

    float acts[7];
    // Row t at dword offset 18*t: 18*t mod 64 injective over a wave32 -> no bank conflicts.
    compute_row(&tile[t * 18], up, dn, wobs, bo0, bo1, wo0, wo1, bo, acts);

    #pragma unroll
    for (int k = 0; k < 7; ++k) otile[t * 7 + k] = acts[k];
    __syncthreads();

    // Coalesced store of 7168 B: 3 x b64 per thread (6144 B) + 1 x b32 (1024 B).
    // Block base = blockIdx*7168 (16B aligned).
    char* ob = (char*)(out + row0 * 7);
    #pragma unroll
    for (int k = 0; k < 3; ++k) {
      const size_t off = (size_t)(t + k * RPB) * 8u;
      *(float2*)(ob + off) = *(const float2*)((const char*)otile + off);
    }
    {
      const size_t off = 6144u + (size_t)t * 4u;
      *(float*)(ob + off) = *(const float*)((const char*)otile + off);
    }
  } else {
    const long long r = row0 + t;
    if (r < (long long)nrows) {
      float rx[18];
      const float* g = x + r * 18;
      #pragma unroll
      for (int i = 0; i < 18; ++i) rx[i] = g[i];
      float acts[7];
      compute_row(rx, up, dn, wobs, bo0, bo1, wo0, wo1, bo, acts);
      float* o = out + r * 7;
      #pragma unroll
      for (int k = 0; k < 7; ++k) o[k] = acts[k];
    }
  }
}

extern "C" void kernel_launch(void* const* d_in, const int* in_sizes, int n_in,
                              void* d_out, int out_size, void* d_ws, size_t ws_size,
                              hipStream_t stream) {
  (void)n_in; (void)out_size; (void)d_ws; (void)ws_size;
  const float* x      = (const float*)d_in[0];
  const float* wihu   = (const float*)d_in[1];
  const float* whhu   = (const float*)d_in[2];
  const float* bihu   = (const float*)d_in[3];
  const float* bhhu   = (const float*)d_in[4];
  const float* wobs   = (const float*)d_in[5];
  const float* bobs   = (const float*)d_in[6];
  const float* wihd   = (const float*)d_in[7];
  const float* whhd   = (const float*)d_in[8];
  const float* bihd   = (const float*)d_in[9];
  const float* bhhd   = (const float*)d_in[10];
  const float* wout   = (const float*)d_in[11];
  const float* boutp  = (const float*)d_in[12];
  float* out = (float*)d_out;

  const int nrows = in_sizes[0] / 18;
  if (nrows <= 0) return;
  const int grid = (nrows + RPB - 1) / RPB;
  rec_policy_kernel<<<grid, RPB, 0, stream>>>(x, wihu, whhu, bihu, bhhu, wobs, bobs,
                                              wihd, whhd, bihd, bhhd, wout, boutp,
                                              out, nrows);
}